// GINViewModel_23536420782504
// MI455X (gfx1250) — compile-verified
//
#include <hip/hip_runtime.h>
#include <math.h>

typedef float f32x2 __attribute__((ext_vector_type(2)));
typedef float f32x8 __attribute__((ext_vector_type(8)));
typedef int   i32x4 __attribute__((vector_size(16)));   // matches builtin proto

#define HID 128
#define OUT_CH 6
#define NUM_GRAPHS 512

#if defined(__HIP_DEVICE_COMPILE__)
typedef __attribute__((address_space(1))) i32x4 gi32x4;  // global-AS int4
typedef __attribute__((address_space(3))) i32x4 li32x4;  // LDS-AS int4
#endif

// ---------------------------------------------------------------- utilities
__global__ __launch_bounds__(256) void copy_f4(const float4* __restrict__ src,
                                               float4* __restrict__ dst, int n4) {
    int i = blockIdx.x * 256 + threadIdx.x;
    if (i < n4) dst[i] = src[i];
}

__global__ __launch_bounds__(256) void zero_f(float* __restrict__ p, int n) {
    int i = blockIdx.x * 256 + threadIdx.x;
    if (i < n) p[i] = 0.0f;
}

// ------------------------------------------------- edge scatter (segment_sum)
// one wave per edge, each lane handles 4 channels (float4 gather + 4 f32 atomics)
__global__ __launch_bounds__(256) void scatter_add(float* __restrict__ acc,
                                                   const float* __restrict__ feat,
                                                   const long long* __restrict__ ei,
                                                   int nEdges) {
    int gid  = blockIdx.x * 256 + threadIdx.x;
    int lane = gid & 31;
    int e    = gid >> 5;
    if (e >= nEdges) return;
    int s = (int)ei[e];
    int d = (int)ei[(size_t)nEdges + e];
    const float4 v = *(const float4*)(feat + (size_t)s * HID + lane * 4);
    float* drow = acc + (size_t)d * HID + lane * 4;
    atomicAdd(drow + 0, v.x);
    atomicAdd(drow + 1, v.y);
    atomicAdd(drow + 2, v.z);
    atomicAdd(drow + 3, v.w);
}

// --------------------------------------------------- fp32 WMMA GEMM (K=N=128)
// C[M,128] = act(A[M,128] @ W[128,128] + bias), act = relu if relu!=0.
// Block = 256 threads = 8 waves; block covers 128 rows; each wave a 16x128 strip.
// Uses V_WMMA_F32_16X16X4_F32 (full fp32 precision, matches reference).
// W is staged into LDS via GLOBAL_LOAD_ASYNC_TO_LDS_B128 (ASYNCcnt path) when
// the toolchain exposes the builtin; otherwise a plain VGPR-staging fallback.
__global__ __launch_bounds__(256) void gemm_wmma_f32(const float* __restrict__ A,
                                                     const float* __restrict__ W,
                                                     const float* __restrict__ bias,
                                                     float* __restrict__ C,
                                                     int M, int relu) {
    extern __shared__ float sW[];  // 128*128 floats = 64 KB
    int t = threadIdx.x;

#if defined(__HIP_DEVICE_COMPILE__) && \
    __has_builtin(__builtin_amdgcn_global_load_async_to_lds_b128)
    {
        // Direct memory->LDS DMA, no VGPR staging. Each lane moves 16 B per
        // instruction; 256 threads x 16 iters covers the 64 KB weight tile.
        const char* gW = (const char*)W;
        char*       lW = (char*)sW;
        for (int off = t * 16; off < HID * HID * (int)sizeof(float); off += 256 * 16) {
            __builtin_amdgcn_global_load_async_to_lds_b128(
                (gi32x4*)(gW + off), (li32x4*)(lW + off), 0, 0);
        }
#if __has_builtin(__builtin_amdgcn_s_wait_asynccnt)
        __builtin_amdgcn_s_wait_asynccnt(0);
#else
        asm volatile("s_wait_asynccnt 0" ::: "memory");
#endif
    }
#else
    for (int i = t; i < HID * HID; i += 256) sW[i] = W[i];
#endif
    __syncthreads();

    int wave    = t >> 5;
    int lane    = t & 31;
    int halfSel = lane >> 4;   // 0: lanes 0-15, 1: lanes 16-31
    int l16     = lane & 15;
    int rowBase = blockIdx.x * 128 + wave * 16;

    // A fragment row for this lane (clamped so EXEC stays all-ones)
    int arow  = rowBase + l16;
    int arowC = (arow < M) ? arow : 0;
    const float* Arow = A + (size_t)arowC * HID;

    // Pull this lane's A row (512 B) toward the caches before the K loop.
    __builtin_prefetch(Arow, 0, 3);

    f32x8 acc[8];
#pragma unroll
    for (int nt = 0; nt < 8; ++nt)
#pragma unroll
        for (int i = 0; i < 8; ++i) acc[nt][i] = 0.0f;

    for (int k = 0; k < HID; k += 4) {
        // A 16x4 fp32 fragment: lanes 0-15 hold K=k,k+1; lanes 16-31 hold K=k+2,k+3
        int koff = k + halfSel * 2;
        f32x2 afrag;
        afrag.x = Arow[koff];
        afrag.y = Arow[koff + 1];
#pragma unroll
        for (int nt = 0; nt < 8; ++nt) {
            // B 4x16 fp32 fragment (symmetric layout): lane -> col N, same K split
            int ncol = nt * 16 + l16;
            f32x2 bfrag;
            bfrag.x = sW[(size_t)koff * HID + ncol];
            bfrag.y = sW[(size_t)(koff + 1) * HID + ncol];
            acc[nt] = __builtin_amdgcn_wmma_f32_16x16x4_f32(
                false, afrag, false, bfrag, (short)0, acc[nt], false, false);
        }
    }

    // C layout: VGPR i -> M = i + 8*halfSel, N = l16 (per 16-col tile)
#pragma unroll
    for (int nt = 0; nt < 8; ++nt) {
        int ncol = nt * 16 + l16;
        float bv = bias[ncol];
#pragma unroll
        for (int i = 0; i < 8; ++i) {
            int row = rowBase + halfSel * 8 + i;
            if (row < M) {
                float v = acc[nt][i] + bv;
                if (relu) v = fmaxf(v, 0.0f);
                C[(size_t)row * HID + ncol] = v;
            }
        }
    }
}

// ------------------------------------------------------------- graph pooling
__global__ __launch_bounds__(256) void pool_add(const float* __restrict__ h,
                                                const long long* __restrict__ batch,
                                                float* __restrict__ pooled,
                                                float* __restrict__ counts,
                                                int nNodes) {
    int gid  = blockIdx.x * 256 + threadIdx.x;
    int lane = gid & 31;
    int n    = gid >> 5;
    if (n >= nNodes) return;
    int g = (int)batch[n];
    const float4 v = *(const float4*)(h + (size_t)n * HID + lane * 4);
    float* p = pooled + (size_t)g * HID + lane * 4;
    atomicAdd(p + 0, v.x);
    atomicAdd(p + 1, v.y);
    atomicAdd(p + 2, v.z);
    atomicAdd(p + 3, v.w);
    if (lane == 0) atomicAdd(counts + g, 1.0f);
}

__global__ __launch_bounds__(256) void pool_div(float* __restrict__ pooled,
                                                const float* __restrict__ counts) {
    int i = blockIdx.x * 256 + threadIdx.x;
    if (i < NUM_GRAPHS * HID) {
        float c = counts[i / HID];
        pooled[i] /= fmaxf(c, 1.0f);
    }
}

// ------------------------------------------------------------- head (128->6)
__global__ __launch_bounds__(256) void final_head(const float* __restrict__ q,
                                                  const float* __restrict__ Wf2,
                                                  const float* __restrict__ bf2,
                                                  float* __restrict__ out) {
    int i = blockIdx.x * 256 + threadIdx.x;
    if (i >= NUM_GRAPHS * OUT_CH) return;
    int g = i / OUT_CH, o = i % OUT_CH;
    float s = bf2[o];
#pragma unroll 4
    for (int k = 0; k < HID; ++k) s += q[(size_t)g * HID + k] * Wf2[k * OUT_CH + o];
    out[i] = 1.0f / (1.0f + expf(-s));
}

// ---------------------------------------------------------------- dispatcher
extern "C" void kernel_launch(void* const* d_in, const int* in_sizes, int n_in,
                              void* d_out, int out_size, void* d_ws, size_t ws_size,
                              hipStream_t stream) {
    const float*     x     = (const float*)d_in[0];
    const long long* ei    = (const long long*)d_in[1];
    const long long* batch = (const long long*)d_in[2];
    const float* W1a = (const float*)d_in[3];
    const float* b1a = (const float*)d_in[4];
    const float* W1b = (const float*)d_in[5];
    const float* b1b = (const float*)d_in[6];
    const float* W2a = (const float*)d_in[7];
    const float* b2a = (const float*)d_in[8];
    const float* W2b = (const float*)d_in[9];
    const float* b2b = (const float*)d_in[10];
    const float* Wf1 = (const float*)d_in[11];
    const float* bf1 = (const float*)d_in[12];
    const float* Wf2 = (const float*)d_in[13];
    const float* bf2 = (const float*)d_in[14];

    const int nNodes = in_sizes[2];
    const int nEdges = in_sizes[1] / 2;

    char* ws = (char*)d_ws;
    const size_t nodeBytes = (size_t)nNodes * HID * sizeof(float);
    float* bufA   = (float*)ws;                       // node features (ping)
    float* bufB   = (float*)(ws + nodeBytes);         // node features (pong)
    float* pooled = (float*)(ws + 2 * nodeBytes);     // [512,128]
    float* counts = pooled + NUM_GRAPHS * HID;        // [512]
    float* q      = counts + NUM_GRAPHS;              // [512,128]

    const int n4 = (nNodes * HID) / 4;
    const int cb = (n4 + 255) / 256;
    const int eb = (nEdges * 32 + 255) / 256;
    const int gb = (nNodes + 127) / 128;
    const int nb = (nNodes * 32 + 255) / 256;
    const size_t shmem = (size_t)HID * HID * sizeof(float);

    // ---- GIN layer 1: t1 = x + segsum(x[src] -> dst); h1 = relu(relu(t1 Wa + ba) Wb + bb)
    copy_f4<<<cb, 256, 0, stream>>>((const float4*)x, (float4*)bufA, n4);
    scatter_add<<<eb, 256, 0, stream>>>(bufA, x, ei, nEdges);
    gemm_wmma_f32<<<gb, 256, shmem, stream>>>(bufA, W1a, b1a, bufB, nNodes, 1);
    gemm_wmma_f32<<<gb, 256, shmem, stream>>>(bufB, W1b, b1b, bufA, nNodes, 1);  // bufA = h1

    // ---- GIN layer 2
    copy_f4<<<cb, 256, 0, stream>>>((const float4*)bufA, (float4*)bufB, n4);
    scatter_add<<<eb, 256, 0, stream>>>(bufB, bufA, ei, nEdges);
    gemm_wmma_f32<<<gb, 256, shmem, stream>>>(bufB, W2a, b2a, bufA, nNodes, 1);
    gemm_wmma_f32<<<gb, 256, shmem, stream>>>(bufA, W2b, b2b, bufB, nNodes, 1);  // bufB = h2

    // ---- global mean pool
    zero_f<<<(NUM_GRAPHS * HID + NUM_GRAPHS + 255) / 256, 256, 0, stream>>>(
        pooled, NUM_GRAPHS * HID + NUM_GRAPHS);
    pool_add<<<nb, 256, 0, stream>>>(bufB, batch, pooled, counts, nNodes);
    pool_div<<<(NUM_GRAPHS * HID + 255) / 256, 256, 0, stream>>>(pooled, counts);

    // ---- head MLP
    gemm_wmma_f32<<<NUM_GRAPHS / 128, 256, shmem, stream>>>(pooled, Wf1, bf1, q, NUM_GRAPHS, 1);
    final_head<<<(NUM_GRAPHS * OUT_CH + 255) / 256, 256, 0, stream>>>(q, Wf2, bf2, (float*)d_out);
}